// ConstrainedSparsemax_13907104105179
// MI455X (gfx1250) — compile-verified
//
#include <hip/hip_runtime.h>
#include <stdint.h>

#define NCOL    8192
#define BLOCK   256
#define VEC     8            // float4 chunks per thread per array
#define PTHREAD (VEC * 4)    // 32 elements per thread per array
#define ITERS   50

// Exact pointee type from clang's diagnostic: int __attribute__((vector_size(16)))
typedef int v4i __attribute__((vector_size(4 * sizeof(int))));
typedef float v4f __attribute__((ext_vector_type(4)));    // native vector for NT stores
typedef __attribute__((address_space(1))) v4i* glb_v4i;   // __device__ (AS1)
typedef __attribute__((address_space(3))) v4i* lds_v4i;   // __shared__ (AS3)

// ---- gfx1250 async global->LDS copy (ASYNCcnt path) ----------------------
__device__ __forceinline__ void async_ld_b128(const void* g, void* l) {
#if __has_builtin(__builtin_amdgcn_global_load_async_to_lds_b128)
  __builtin_amdgcn_global_load_async_to_lds_b128(
      (glb_v4i)g, (lds_v4i)l, /*offset=*/0, /*cpol=*/0);
#else
  unsigned loff = (unsigned)(uintptr_t)(__attribute__((address_space(3))) void*)l;
  asm volatile("global_load_async_to_lds_b128 %0, %1, off"
               :: "v"(loff), "v"(g) : "memory");
#endif
}

__device__ __forceinline__ void wait_async0() {
#if __has_builtin(__builtin_amdgcn_s_wait_asynccnt)
  __builtin_amdgcn_s_wait_asynccnt(0);
#else
  asm volatile("s_wait_asynccnt 0" ::: "memory");
#endif
}

// clip(x, 0, u) in one VALU op (valid since u >= 0): v_med3_f32
__device__ __forceinline__ float clip0u(float x, float u) {
  return __builtin_amdgcn_fmed3f(x, 0.0f, u);
}

// ---- wave32 reductions (explicit width 32: CDNA5 is wave32-only) ---------
__device__ __forceinline__ float wave_sum(float v) {
#pragma unroll
  for (int m = 16; m > 0; m >>= 1) v += __shfl_xor(v, m, 32);
  return v;
}
__device__ __forceinline__ float wave_min(float v) {
#pragma unroll
  for (int m = 16; m > 0; m >>= 1) v = fminf(v, __shfl_xor(v, m, 32));
  return v;
}
__device__ __forceinline__ float wave_max(float v) {
#pragma unroll
  for (int m = 16; m > 0; m >>= 1) v = fmaxf(v, __shfl_xor(v, m, 32));
  return v;
}

__global__ __launch_bounds__(BLOCK)
void csparsemax_kernel(const float* __restrict__ zg,
                       const float* __restrict__ ug,
                       float* __restrict__ og) {
  extern __shared__ char dynsmem[];
  v4f* zs = (v4f*)dynsmem;                // 32 KB: one row of z
  v4f* us = zs + (NCOL / 4);              // 32 KB: one row of u
  __shared__ float red[4][8];             // cross-wave reduction slots

  const int row  = blockIdx.x;
  const int t    = threadIdx.x;
  const int wid  = t >> 5;
  const int lane = t & 31;

  const v4f* zrow = (const v4f*)(zg + (size_t)row * NCOL);
  const v4f* urow = (const v4f*)(ug + (size_t)row * NCOL);

  // ---- stage the full row HBM -> LDS via async DMA (no VGPR return path) --
#pragma unroll
  for (int j = 0; j < VEC; ++j) {
    const int idx = j * BLOCK + t;        // fully coalesced b128 per lane
    async_ld_b128(zrow + idx, zs + idx);
    async_ld_b128(urow + idx, us + idx);
  }
  wait_async0();  // each lane reads back exactly what it issued -> no barrier

  // ---- LDS -> registers once; bisection loop is register-resident ---------
  float z[PTHREAD], u[PTHREAD];
#pragma unroll
  for (int j = 0; j < VEC; ++j) {
    const int idx = j * BLOCK + t;
    *(v4f*)&z[4 * j] = zs[idx];
    *(v4f*)&u[4 * j] = us[idx];
  }

  // lo = min(z-u), hi = max(z) over the row
  float lmin = 3.0e38f, lmax = -3.0e38f;
#pragma unroll
  for (int i = 0; i < PTHREAD; ++i) {
    lmin = fminf(lmin, z[i] - u[i]);
    lmax = fmaxf(lmax, z[i]);
  }
  lmin = wave_min(lmin);
  lmax = wave_max(lmax);
  if (lane == 0) { red[0][wid] = lmin; red[1][wid] = lmax; }
  __syncthreads();
  float lo = red[0][0], hi = red[1][0];
#pragma unroll
  for (int k = 1; k < 8; ++k) { lo = fminf(lo, red[0][k]); hi = fmaxf(hi, red[1][k]); }
  __syncthreads();

  // ---- 50 bisection iterations; 1 barrier/iter via parity double-buffer ---
  for (int it = 0; it < ITERS; ++it) {
    const float mid = 0.5f * (lo + hi);
    float s = 0.0f;
#pragma unroll
    for (int i = 0; i < PTHREAD; ++i)
      s += clip0u(z[i] - mid, u[i]);      // v_sub + v_med3 + v_add
    s = wave_sum(s);
    if (lane == 0) red[it & 1][wid] = s;
    __syncthreads();
    float f = 0.0f;
#pragma unroll
    for (int k = 0; k < 8; ++k) f += red[it & 1][k];
    if (f > 1.0f) lo = mid; else hi = mid;
  }
  __syncthreads();

  // ---- exact tau from identified active set -------------------------------
  const float tau0 = 0.5f * (lo + hi);
  float cnt = 0.0f, s1 = 0.0f, s2 = 0.0f;
#pragma unroll
  for (int i = 0; i < PTHREAD; ++i) {
    const float d = z[i] - tau0;
    const bool r1 = (d > 0.0f) & (d < u[i]);
    const bool r2 = (d >= u[i]);
    cnt += r1 ? 1.0f : 0.0f;
    s1  += r1 ? z[i] : 0.0f;
    s2  += r2 ? u[i] : 0.0f;
  }
  cnt = wave_sum(cnt);
  s1  = wave_sum(s1);
  s2  = wave_sum(s2);
  if (lane == 0) { red[0][wid] = cnt; red[1][wid] = s1; red[2][wid] = s2; }
  __syncthreads();
  float C = 0.0f, S1 = 0.0f, S2 = 0.0f;
#pragma unroll
  for (int k = 0; k < 8; ++k) { C += red[0][k]; S1 += red[1][k]; S2 += red[2][k]; }
  float tau = (S1 + S2 - 1.0f) / fmaxf(C, 1.0f);
  tau = (C > 0.0f) ? tau : tau0;

  // ---- probs = r1*(z - tau) + r2*u; streaming (non-temporal) b128 stores --
  v4f* orow = (v4f*)(og + (size_t)row * NCOL);
#pragma unroll
  for (int j = 0; j < VEC; ++j) {
    v4f o;
#pragma unroll
    for (int c = 0; c < 4; ++c) {
      const int i = 4 * j + c;
      const float d0 = z[i] - tau0;
      const bool r1 = (d0 > 0.0f) & (d0 < u[i]);
      const bool r2 = (d0 >= u[i]);
      o[c] = r1 ? (z[i] - tau) : (r2 ? u[i] : 0.0f);
    }
    __builtin_nontemporal_store(o, orow + j * BLOCK + t);
  }
}

extern "C" void kernel_launch(void* const* d_in, const int* in_sizes, int n_in,
                              void* d_out, int out_size, void* d_ws, size_t ws_size,
                              hipStream_t stream) {
  (void)n_in; (void)d_ws; (void)ws_size; (void)in_sizes;
  const float* z = (const float*)d_in[0];
  const float* u = (const float*)d_in[1];
  float* out = (float*)d_out;

  const int rows = out_size / NCOL;                       // 4096
  const size_t shmem = 2 * (size_t)NCOL * sizeof(float);  // 64 KB/row staging
  csparsemax_kernel<<<rows, BLOCK, shmem, stream>>>(z, u, out);
}